// MultiVariateGaussian_12910671692206
// MI455X (gfx1250) — compile-verified
//
#include <hip/hip_runtime.h>

#define BB   32      // batch
#define CC   256     // channels
#define NN   4096    // H*W locations
#define NB   16      // n-locations per workgroup
#define ROWP 20      // LDS row pitch in floats: 16 n + 4 pad (80B, 16B aligned)
#define TPB  256     // 8 waves of 32
#define REGV 0.01f

typedef __attribute__((ext_vector_type(2))) float v2f;
typedef __attribute__((ext_vector_type(8))) float v8f;
typedef __attribute__((ext_vector_type(4))) int   v4i;

typedef __attribute__((address_space(1))) v4i gv4i;   // global 16B chunk
typedef __attribute__((address_space(3))) v4i lv4i;   // LDS 16B chunk

#if defined(__has_builtin)
#if __has_builtin(__builtin_amdgcn_global_load_async_to_lds_b128)
#define HAVE_ASYNC_LDS 1
#endif
#endif
#ifndef HAVE_ASYNC_LDS
#define HAVE_ASYNC_LDS 0
#endif

// Wait until at most N async global->LDS loads from this wave are outstanding.
// Async loads complete in order, so N=8 releases the older of two 8-load batches.
template <int N>
__device__ __forceinline__ void wait_async_lds() {
#if HAVE_ASYNC_LDS
#if __has_builtin(__builtin_amdgcn_s_wait_asynccnt)
  __builtin_amdgcn_s_wait_asynccnt(N);
#else
  asm volatile("s_wait_asynccnt %0" ::"i"(N) : "memory");
#endif
#endif
}

// Copy raw x[b, t0+cl, n0 .. n0+NB) -> lds[(b*16+cl)*ROWP + nl], 16B chunks.
// gsrc is pre-offset to (t0, n0). Pure copy: 8 async b128 loads per wave.
__device__ __forceinline__ void stage_tile(const float* __restrict__ gsrc,
                                           float* lds, int tid) {
  for (int i = tid; i < BB * 16 * (NB / 4); i += TPB) {
    const int nq = i & 3;
    const int cl = (i >> 2) & 15;
    const int b  = i >> 6;
    const float* src = gsrc + (size_t)(b * CC + cl) * NN + nq * 4;
    float*       dst = lds + (b * 16 + cl) * ROWP + nq * 4;
#if HAVE_ASYNC_LDS
    __builtin_amdgcn_global_load_async_to_lds_b128((gv4i*)src, (lv4i*)dst, 0, 0);
#else
    *(float4*)dst = *(const float4*)src;
#endif
  }
}

// ---------------- mean over batch: mean[c,n] = (1/B) sum_b x[b,c,n] ----------
__global__ __launch_bounds__(TPB) void mvg_mean_kernel(const float* __restrict__ x,
                                                       float* __restrict__ mean) {
  const size_t i = (size_t)blockIdx.x * TPB + threadIdx.x;   // over C*N (exact multiple)
  float s = 0.f;
#pragma unroll
  for (int b = 0; b < BB; ++b) s += x[(size_t)b * CC * NN + i];
  mean[i] = s * (1.0f / BB);
}

// ------------- per-location Gram via V_WMMA_F32_16X16X4_F32 ------------------
// De-centered: cov[c,d,n] = (sum_b x_bc x_bd - B*mean_c*mean_d)/(B-1) + REG*I.
// Workgroup = (cTile of 16 channels) x (16 consecutive n). Raw x tiles staged
// with async global->LDS b128 DMA, double-buffered so tile dt+1 streams in
// while tile dt is consumed by the fp32 WMMA Gram chain.
__global__ __launch_bounds__(TPB) void mvg_cov_kernel(const float* __restrict__ x,
                                                      const float* __restrict__ mean,
                                                      float* __restrict__ cov) {
  __shared__ float As[BB * 16 * ROWP];        // 40 KB
  __shared__ float Bs[2][BB * 16 * ROWP];     // 2 x 40 KB (double buffer)

  const int ct   = blockIdx.x & 15;   // cTile
  const int nb   = blockIdx.x >> 4;   // n-block
  const int n0   = nb * NB;
  const int c0   = ct * 16;
  const int tid  = threadIdx.x;
  const int lane = tid & 31;
  const int wave = tid >> 5;          // 0..7
  const int m    = lane & 15;         // M/N lane index inside 16x16 tile
  const int hi   = lane >> 4;         // half-wave
  const int koff = hi * 2;            // K offset for A/B fragments

  stage_tile(x + (size_t)c0 * NN + n0, As, tid);      // 8 async loads / wave
  stage_tile(x + (size_t)0 * NN + n0, Bs[0], tid);    // 8 more (dTile 0)
  wait_async_lds<8>();                                // As complete
  __syncthreads();

  // Hoist A fragments (8 K-steps, K=4 covers b = 4*k0..4*k0+3), the mean_c
  // epilogue terms, and the lane-constant diagonal increments.
  v2f   afr[2][8];
  float mcs[2][8];
  float diagAdd[8];
#pragma unroll
  for (int r = 0; r < 2; ++r) {
    const int nl = wave + r * 8;
#pragma unroll
    for (int k0 = 0; k0 < 8; ++k0) {
      afr[r][k0].x = As[((k0 * 4 + koff + 0) * 16 + m) * ROWP + nl];
      afr[r][k0].y = As[((k0 * 4 + koff + 1) * 16 + m) * ROWP + nl];
    }
#pragma unroll
    for (int g = 0; g < 8; ++g) {     // D layout: VGPR g -> row g (lanes 0-15) / g+8
      const int row = g + hi * 8;
      mcs[r][g] = mean[(size_t)(c0 + row) * NN + (n0 + nl)] * ((float)BB / (BB - 1));
    }
  }
#pragma unroll
  for (int g = 0; g < 8; ++g) diagAdd[g] = ((g + hi * 8) == m) ? REGV : 0.0f;

  for (int dt = 0; dt < 16; ++dt) {
    const int d0  = dt * 16;
    const int cur = dt & 1;
    __syncthreads();                  // all waves done reading buffer cur^1
    if (dt < 15) {
      stage_tile(x + (size_t)(d0 + 16) * NN + n0, Bs[cur ^ 1], tid);
      wait_async_lds<8>();            // buffer `cur` (older batch) complete
    } else {
      wait_async_lds<0>();
    }
    __syncthreads();                  // buffer `cur` visible workgroup-wide

#pragma unroll
    for (int r = 0; r < 2; ++r) {
      const int nl = wave + r * 8;
      const float md = mean[(size_t)(d0 + m) * NN + (n0 + nl)];

      // Prefetch all 8 B fragments so the 8 WMMAs issue back-to-back.
      v2f bf[8];
#pragma unroll
      for (int k0 = 0; k0 < 8; ++k0) {
        bf[k0].x = Bs[cur][((k0 * 4 + koff + 0) * 16 + m) * ROWP + nl];
        bf[k0].y = Bs[cur][((k0 * 4 + koff + 1) * 16 + m) * ROWP + nl];
      }

      v8f acc = {};
#pragma unroll
      for (int k0 = 0; k0 < 8; ++k0) {
        // D(16x16 f32) += A(16x4 f32) * B(4x16 f32), K = 4 batch samples
        acc = __builtin_amdgcn_wmma_f32_16x16x4_f32(false, afr[r][k0], false, bf[k0],
                                                    (short)0, acc, false, false);
      }

      const float* ap = (const float*)&acc;
#pragma unroll
      for (int g = 0; g < 8; ++g) {
        const int c = c0 + g + hi * 8;
        const int d = d0 + m;
        float v = ap[g] * (1.0f / (BB - 1)) - mcs[r][g] * md;
        if (dt == ct) v += diagAdd[g];       // scalar-gated diagonal REG
        cov[((size_t)c * CC + d) * NN + (size_t)(n0 + nl)] = v;
      }
    }
  }
}

extern "C" void kernel_launch(void* const* d_in, const int* in_sizes, int n_in,
                              void* d_out, int out_size, void* d_ws, size_t ws_size,
                              hipStream_t stream) {
  (void)in_sizes; (void)n_in; (void)out_size; (void)d_ws; (void)ws_size;
  const float* x = (const float*)d_in[0];
  float* out     = (float*)d_out;
  float* mean    = out;                      // [C, N]
  float* cov     = out + (size_t)CC * NN;    // [C, C, N]

  mvg_mean_kernel<<<(CC * NN) / TPB, TPB, 0, stream>>>(x, mean);
  // 16 cTiles * (4096/16) n-blocks = 4096 workgroups
  mvg_cov_kernel<<<(NN / NB) * 16, TPB, 0, stream>>>(x, mean, cov);
}